// PT_MAP_27960237097565
// MI455X (gfx1250) — compile-verified
//
#include <hip/hip_runtime.h>
#include <math.h>

#define NUM_WAY   32
#define NUM_SHOT  5
#define NUM_QUERY 40
#define DIMS      16384
#define N_L       160
#define N_U       1280
#define N_TOT     1440
#define LAM       10.0f
#define ALPHA     0.2f
#define N_EPOCHS  20
#define SINK_MAXITERS 50
#define SINK_EPS  1e-6f

typedef __attribute__((ext_vector_type(2))) float v2f;
typedef __attribute__((ext_vector_type(8))) float v8f;
typedef __attribute__((ext_vector_type(4))) unsigned int v4u;
typedef __attribute__((ext_vector_type(8))) int v8i_t;
typedef __attribute__((ext_vector_type(4))) int v4i_t;
typedef __attribute__((address_space(3))) float lds_f;

// ---------- small helpers ----------
__device__ __forceinline__ float wred_sum(float v) {
  #pragma unroll
  for (int off = 16; off > 0; off >>= 1) v += __shfl_xor(v, off, 32);
  return v;
}
__device__ __forceinline__ float wred_max(float v) {
  #pragma unroll
  for (int off = 16; off > 0; off >>= 1) v = fmaxf(v, __shfl_xor(v, off, 32));
  return v;
}

// ---------- preprocessing ----------
// inv1[i] = 1 / max(||sqrt(X[i]+eps)||, 1e-12)
__global__ __launch_bounds__(256)
void k_rownorm1(const float* __restrict__ X, float* __restrict__ inv1) {
  __shared__ float red[256];
  const int i = blockIdx.x;
  float s = 0.f;
  for (int d = threadIdx.x; d < DIMS; d += 256)
    s += X[(size_t)i * DIMS + d] + 1e-6f;             // ||sqrt(x+eps)||^2 = sum(x+eps)
  red[threadIdx.x] = s; __syncthreads();
  for (int st = 128; st > 0; st >>= 1) { if (threadIdx.x < st) red[threadIdx.x] += red[threadIdx.x + st]; __syncthreads(); }
  if (threadIdx.x == 0) inv1[i] = 1.0f / fmaxf(sqrtf(red[0]), 1e-12f);
}

// column means over support rows and query rows of the row-normalized data
__global__ __launch_bounds__(256)
void k_colmean(const float* __restrict__ X, const float* __restrict__ inv1,
               float* __restrict__ mean_s, float* __restrict__ mean_q) {
  const int d = blockIdx.x * 256 + threadIdx.x;
  float ss = 0.f, sq = 0.f;
  for (int i = 0; i < N_L; ++i)
    ss += sqrtf(X[(size_t)i * DIMS + d] + 1e-6f) * inv1[i];
  for (int i = N_L; i < N_TOT; ++i)
    sq += sqrtf(X[(size_t)i * DIMS + d] + 1e-6f) * inv1[i];
  mean_s[d] = ss * (1.0f / N_L);
  mean_q[d] = sq * (1.0f / N_U);
}

// Z[i] = l2norm( rownorm(sqrt(X[i]+eps)) - mean_block )
__global__ __launch_bounds__(256)
void k_center(const float* __restrict__ X, const float* __restrict__ inv1,
              const float* __restrict__ mean_s, const float* __restrict__ mean_q,
              float* __restrict__ Z) {
  __shared__ float red[256];
  const int i = blockIdx.x;
  const float* mean = (i < N_L) ? mean_s : mean_q;
  const float inv = inv1[i];
  float s2 = 0.f;
  for (int d = threadIdx.x; d < DIMS; d += 256) {
    float t = sqrtf(X[(size_t)i * DIMS + d] + 1e-6f) * inv - mean[d];
    s2 = fmaf(t, t, s2);
  }
  red[threadIdx.x] = s2; __syncthreads();
  for (int st = 128; st > 0; st >>= 1) { if (threadIdx.x < st) red[threadIdx.x] += red[threadIdx.x + st]; __syncthreads(); }
  const float invn = 1.0f / fmaxf(sqrtf(red[0]), 1e-12f);
  for (int d = threadIdx.x; d < DIMS; d += 256) {
    float t = sqrtf(X[(size_t)i * DIMS + d] + 1e-6f) * inv - mean[d];
    Z[(size_t)i * DIMS + d] = t * invn;
  }
}

// zz[n] = ||Z_q[n]||^2 (exactly as the reference computes it; ~1 but not assumed)
__global__ __launch_bounds__(256)
void k_zz(const float* __restrict__ Zq, float* __restrict__ zz) {
  __shared__ float red[256];
  const int n = blockIdx.x;
  float s = 0.f;
  for (int d = threadIdx.x; d < DIMS; d += 256) {
    float v = Zq[(size_t)n * DIMS + d];
    s = fmaf(v, v, s);
  }
  red[threadIdx.x] = s; __syncthreads();
  for (int st = 128; st > 0; st >>= 1) { if (threadIdx.x < st) red[threadIdx.x] += red[threadIdx.x + st]; __syncthreads(); }
  if (threadIdx.x == 0) zz[n] = red[0];
}

// mus[k,d] = mean over the 5 shots (support row layout: shot*32 + way)
__global__ __launch_bounds__(256)
void k_mus_init(const float* __restrict__ Z, float* __restrict__ mus) {
  const int idx = blockIdx.x * 256 + threadIdx.x;   // 32*16384 threads
  const int d = idx & (DIMS - 1);
  const int k = idx >> 14;
  float s = 0.f;
  #pragma unroll
  for (int sh = 0; sh < NUM_SHOT; ++sh)
    s += Z[(size_t)(sh * NUM_WAY + k) * DIMS + d];
  mus[(size_t)k * DIMS + d] = s * (1.0f / NUM_SHOT);
}

// one-hot support rows of P (written once)
__global__ __launch_bounds__(256)
void k_onehot(float* __restrict__ P) {
  const int i = blockIdx.x * 256 + threadIdx.x;     // < 160*32
  if (i < N_L * NUM_WAY) {
    const int n = i >> 5, k = i & 31;
    P[i] = (k == (n & 31)) ? 1.0f : 0.0f;
  }
}

// mm[k] = ||mus[k]||^2
__global__ __launch_bounds__(256)
void k_mm(const float* __restrict__ mus, float* __restrict__ mm) {
  __shared__ float red[256];
  const int k = blockIdx.x;
  float s = 0.f;
  for (int d = threadIdx.x; d < DIMS; d += 256) {
    float v = mus[(size_t)k * DIMS + d];
    s = fmaf(v, v, s);
  }
  red[threadIdx.x] = s; __syncthreads();
  for (int st = 128; st > 0; st >>= 1) { if (threadIdx.x < st) red[threadIdx.x] += red[threadIdx.x + st]; __syncthreads(); }
  if (threadIdx.x == 0) mm[k] = red[0];
}

__global__ __launch_bounds__(256)
void k_zero(float* __restrict__ p, int n) {
  const int i = blockIdx.x * 256 + threadIdx.x;
  if (i < n) p[i] = 0.0f;
}

// ---------- cross terms cr[n,k] = Z_q[n] . mus[k] via V_WMMA_F32_16X16X4_F32 ----------
// 1 wave per (16x16 tile, K-chunk). Grid = 80 n-tiles * 2 k-tiles * 8 K-chunks = 1280 waves.
// A fragment (16x4 f32): lane<16 -> row M=lane, v2f = {K0,K1}; lane>=16 -> {K2,K3}.
// B fragment (4x16 f32): symmetric (col N = lane&15, K half selected by lane>>4).
// C/D (16x16 f32): v[j] : M = j + 8*(lane>>4), N = lane&15.
__global__ __launch_bounds__(32)
void k_dist_wmma(const float* __restrict__ Zq, const float* __restrict__ mus,
                 float* __restrict__ cr) {
  const int bid   = blockIdx.x;
  const int nt    = bid % 80;
  const int kt    = (bid / 80) & 1;
  const int chunk = bid / 160;                 // 0..7, each covers K span of 2048
  const int lane  = threadIdx.x;
  const int half  = lane >> 4;
  const int l15   = lane & 15;

  const float* A = Zq  + (size_t)(nt * 16 + l15) * DIMS;
  const float* B = mus + (size_t)(kt * 16 + l15) * DIMS;

  v8f c = {0.f, 0.f, 0.f, 0.f, 0.f, 0.f, 0.f, 0.f};
  const int kbeg = chunk * 2048;
  #pragma unroll 8
  for (int kk = kbeg; kk < kbeg + 2048; kk += 4) {
    const int o = kk + half * 2;
    v2f a = *(const v2f*)(A + o);
    v2f b = *(const v2f*)(B + o);
    c = __builtin_amdgcn_wmma_f32_16x16x4_f32(false, a, false, b, (short)0, c, false, false);
  }
  #pragma unroll
  for (int j = 0; j < 8; ++j) {
    const int row = nt * 16 + half * 8 + j;    // query index
    const int col = kt * 16 + l15;             // way index
    atomicAdd(&cr[row * NUM_WAY + col], c[j]);
  }
}

// dist[n,k] = max(zz[n] + mm[k] - 2*cr, 0)
__global__ __launch_bounds__(256)
void k_dist_finish(const float* __restrict__ cr, const float* __restrict__ mm,
                   const float* __restrict__ zz, float* __restrict__ dist) {
  const int i = blockIdx.x * 256 + threadIdx.x;  // < 1280*32
  if (i < N_U * NUM_WAY) {
    const int k = i & 31;
    const int n = i >> 5;
    dist[i] = fmaxf(zz[n] + mm[k] - 2.0f * cr[i], 0.0f);
  }
}

// ---------- Sinkhorn (single workgroup, 640 threads = 20 waves, 64 rows/wave) ----------
__global__ __launch_bounds__(640)
void k_sinkhorn(const float* __restrict__ dist, float* __restrict__ P) {
  __shared__ float u_s[N_U];
  __shared__ float s_s[N_U];
  __shared__ float red[32];
  __shared__ float colp[20][NUM_WAY];
  __shared__ float colscale[NUM_WAY];
  __shared__ float gmax_s;

  const int t = threadIdx.x, wave = t >> 5, lane = t & 31;
  const int r0 = wave * 64, r1 = r0 + 64;

  // P = exp(-lam*dist); global-normalize
  float part = 0.f;
  for (int n = r0; n < r1; ++n) {
    float v = __expf(-LAM * dist[n * NUM_WAY + lane]);
    P[n * NUM_WAY + lane] = v;
    part += v;
  }
  part = wred_sum(part);
  if (lane == 0) red[wave] = part;
  __syncthreads();
  if (t == 0) {
    float tot = 0.f;
    for (int w = 0; w < 20; ++w) tot += red[w];
    red[0] = 1.0f / tot;
  }
  __syncthreads();
  const float invtot = red[0];
  for (int n = r0; n < r1; ++n) P[n * NUM_WAY + lane] *= invtot;
  u_s[2 * t] = 0.f; u_s[2 * t + 1] = 0.f;
  __syncthreads();

  for (int it = 0; it < SINK_MAXITERS; ++it) {
    // row sums
    for (int n = r0; n < r1; ++n) {
      float s = wred_sum(P[n * NUM_WAY + lane]);
      if (lane == 0) s_s[n] = s;
    }
    __syncthreads();
    // convergence: max |u - s|
    float m = fmaxf(fabsf(u_s[2 * t] - s_s[2 * t]), fabsf(u_s[2 * t + 1] - s_s[2 * t + 1]));
    m = wred_max(m);
    if (lane == 0) red[wave] = m;
    __syncthreads();
    if (t == 0) {
      float g = 0.f;
      for (int w = 0; w < 20; ++w) g = fmaxf(g, red[w]);
      gmax_s = g;
    }
    __syncthreads();
    if (gmax_s <= SINK_EPS) break;   // uniform; identical to reference's frozen updates
    // column sums of row-rescaled P (scaling folded into the read)
    float cp = 0.f;
    for (int n = r0; n < r1; ++n) cp += P[n * NUM_WAY + lane] / s_s[n];
    colp[wave][lane] = cp;
    __syncthreads();
    if (t < NUM_WAY) {
      float cs = 0.f;
      for (int w = 0; w < 20; ++w) cs += colp[w][t];
      colscale[t] = (float)NUM_QUERY / cs;
    }
    __syncthreads();
    const float csl = colscale[lane];
    for (int n = r0; n < r1; ++n)
      P[n * NUM_WAY + lane] = P[n * NUM_WAY + lane] * (1.0f / s_s[n]) * csl;
    u_s[2 * t] = s_s[2 * t]; u_s[2 * t + 1] = s_s[2 * t + 1];
    __syncthreads();
  }
}

// colsum[k] = sum_n P[n,k] over all 1440 rows (support onehot + queries)
__global__ __launch_bounds__(1024)
void k_colsum(const float* __restrict__ P, float* __restrict__ colsum) {
  __shared__ float part[32][NUM_WAY];
  const int t = threadIdx.x, w = t >> 5, l = t & 31;
  float cp = 0.f;
  for (int n = w * 45; n < w * 45 + 45; ++n) cp += P[n * NUM_WAY + l];
  part[w][l] = cp; __syncthreads();
  if (t < NUM_WAY) {
    float s = 0.f;
    for (int i = 0; i < 32; ++i) s += part[i][t];
    colsum[t] = s;
  }
}

// mus[k,d] += alpha * (sum_n P[n,k] Z[n,d] / colsum[k] - mus[k,d])
// P chunks (96 rows x 32 = 12 KB, contiguous) are staged into LDS by the
// Tensor Data Mover: wave 0 issues TENSOR_LOAD_TO_LDS with a 1-D D# and
// waits on TENSORcnt; the block barrier releases the consumers.
__global__ __launch_bounds__(128)
void k_emus(const float* __restrict__ Z, const float* __restrict__ P,
            const float* __restrict__ colsum, float* __restrict__ mus) {
  __shared__ float Pl[96][NUM_WAY];   // 12 KB chunk of p (TDM destination)
  __shared__ float csl[NUM_WAY];
  const int tid = threadIdx.x;
  const int d = blockIdx.x * 128 + tid;
  if (tid < NUM_WAY) csl[tid] = colsum[tid];
  float acc[NUM_WAY];
  #pragma unroll
  for (int k = 0; k < NUM_WAY; ++k) acc[k] = 0.f;

  const unsigned lds_base =
      (unsigned)(unsigned long long)(lds_f*)&Pl[0][0];   // LDS byte offset of Pl

  for (int n0 = 0; n0 < N_TOT; n0 += 96) {          // 15 chunks of 96 rows
    __syncthreads();                                // previous chunk fully consumed
    if (tid < 32) {                                 // wave 0 drives the TDM
      const unsigned long long ga =
          (unsigned long long)(const void*)(P + (size_t)n0 * NUM_WAY);
      const unsigned ne = 96u * NUM_WAY;            // 3072 f32 elements, 1-D tile
      v4u g0;
      g0.x = 1u;                                            // count=1, user mode
      g0.y = lds_base;                                      // lds_addr
      g0.z = (unsigned)(ga & 0xffffffffu);                  // global_addr[31:0]
      g0.w = (unsigned)((ga >> 32) & 0x1ffffffu) | (2u << 30); // addr[56:32] | type=2
      v8i_t g1;
      g1[0] = (int)(2u << 16);                              // data_size=4B; no mask/pad
      g1[1] = (int)((ne & 0xffffu) << 16);                  // tensor_dim0[15:0]
      g1[2] = (int)((ne >> 16) | (1u << 16));               // tensor_dim0[31:16] | tensor_dim1=1
      g1[3] = (int)(ne << 16);                              // tile_dim0 = 3072
      g1[4] = 1;                                            // tile_dim1 = 1, tile_dim2 = 0
      g1[5] = (int)ne;                                      // tensor_dim0_stride
      g1[6] = 0;
      g1[7] = 0;
      const v4i_t gz4 = {0, 0, 0, 0};                       // groups 2/3 unused (<=2-D)
      const v8i_t gz8 = {0, 0, 0, 0, 0, 0, 0, 0};           // extra group (clang-23 form)
      __builtin_amdgcn_tensor_load_to_lds(g0, g1, gz4, gz4, gz8, 0);
      __builtin_amdgcn_s_wait_tensorcnt(0);
    }
    __syncthreads();
    for (int nl = 0; nl < 96; ++nl) {
      const float z = Z[(size_t)(n0 + nl) * DIMS + d];
      #pragma unroll
      for (int k = 0; k < NUM_WAY; ++k) acc[k] = fmaf(Pl[nl][k], z, acc[k]);
    }
  }
  #pragma unroll
  for (int k = 0; k < NUM_WAY; ++k) {
    const size_t idx = (size_t)k * DIMS + d;
    const float m = mus[idx];
    mus[idx] = m + ALPHA * (acc[k] / csl[k] - m);
  }
}

// acc over queries; labels[n] = n % 32 by construction
__global__ __launch_bounds__(1024)
void k_finalize(const float* __restrict__ P, float* __restrict__ out) {
  __shared__ int red[1024];
  const int t = threadIdx.x;
  int cnt = 0;
  for (int idx = t; idx < N_U; idx += 1024) {
    const float* row = P + (size_t)(N_L + idx) * NUM_WAY;
    float best = row[0]; int bi = 0;
    #pragma unroll
    for (int k = 1; k < NUM_WAY; ++k) { if (row[k] > best) { best = row[k]; bi = k; } }
    cnt += (bi == ((N_L + idx) & 31)) ? 1 : 0;
  }
  red[t] = cnt; __syncthreads();
  for (int st = 512; st > 0; st >>= 1) { if (t < st) red[t] += red[t + st]; __syncthreads(); }
  if (t == 0) out[0] = (float)red[0] / (float)N_U;
}

__global__ __launch_bounds__(256)
void k_copyP(const float* __restrict__ P, float* __restrict__ out, int n) {
  const int i = blockIdx.x * 256 + threadIdx.x;
  if (i < n) out[1 + i] = P[i];
}

// ---------- host orchestration ----------
extern "C" void kernel_launch(void* const* d_in, const int* in_sizes, int n_in,
                              void* d_out, int out_size, void* d_ws, size_t ws_size,
                              hipStream_t stream) {
  (void)in_sizes; (void)n_in; (void)ws_size;
  const float* X = (const float*)d_in[0];
  float* out = (float*)d_out;

  char* ws = (char*)d_ws;
  size_t off = 0;
  auto alloc = [&](size_t bytes) -> float* {
    float* p = (float*)(ws + off);
    off = (off + bytes + 255) & ~(size_t)255;
    return p;
  };
  float* Z      = alloc((size_t)N_TOT * DIMS * 4);   // 94.4 MB (L2-resident)
  float* mus    = alloc((size_t)NUM_WAY * DIMS * 4);
  float* inv1   = alloc((size_t)N_TOT * 4);
  float* mean_s = alloc((size_t)DIMS * 4);
  float* mean_q = alloc((size_t)DIMS * 4);
  float* mm     = alloc((size_t)NUM_WAY * 4);
  float* colsum = alloc((size_t)NUM_WAY * 4);
  float* zz     = alloc((size_t)N_U * 4);
  float* cr     = alloc((size_t)N_U * NUM_WAY * 4);
  float* dist   = alloc((size_t)N_U * NUM_WAY * 4);
  float* P      = alloc((size_t)N_TOT * NUM_WAY * 4);

  const float* Zq = Z + (size_t)N_L * DIMS;
  float* Pq = P + (size_t)N_L * NUM_WAY;

  // preprocessing
  k_rownorm1<<<N_TOT, 256, 0, stream>>>(X, inv1);
  k_colmean<<<DIMS / 256, 256, 0, stream>>>(X, inv1, mean_s, mean_q);
  k_center<<<N_TOT, 256, 0, stream>>>(X, inv1, mean_s, mean_q, Z);
  k_zz<<<N_U, 256, 0, stream>>>(Zq, zz);
  k_mus_init<<<(NUM_WAY * DIMS) / 256, 256, 0, stream>>>(Z, mus);
  k_onehot<<<(N_L * NUM_WAY + 255) / 256, 256, 0, stream>>>(P);

  // 20 update epochs + 1 final probas pass
  for (int e = 0; e <= N_EPOCHS; ++e) {
    k_mm<<<NUM_WAY, 256, 0, stream>>>(mus, mm);
    k_zero<<<(N_U * NUM_WAY + 255) / 256, 256, 0, stream>>>(cr, N_U * NUM_WAY);
    k_dist_wmma<<<80 * 2 * 8, 32, 0, stream>>>(Zq, mus, cr);
    k_dist_finish<<<(N_U * NUM_WAY + 255) / 256, 256, 0, stream>>>(cr, mm, zz, dist);
    k_sinkhorn<<<1, 640, 0, stream>>>(dist, Pq);
    if (e < N_EPOCHS) {
      k_colsum<<<1, 1024, 0, stream>>>(P, colsum);
      k_emus<<<DIMS / 128, 128, 0, stream>>>(Z, P, colsum, mus);
    }
  }

  // outputs: [acc, P(1440x32)]
  k_finalize<<<1, 1024, 0, stream>>>(P, out);
  const int np = N_TOT * NUM_WAY;
  const int ncopy = (out_size - 1 < np) ? (out_size - 1) : np;
  k_copyP<<<(ncopy + 255) / 256, 256, 0, stream>>>(P, out, ncopy);
}